// JosephProjector3D_38216619000403
// MI455X (gfx1250) — compile-verified
//
#include <hip/hip_runtime.h>
#include <math.h>

#define A_N 144
#define V_N 48
#define U_N 192
#define D_N 48
#define H_N 192
#define W_N 192
#define NSTEPS 192
#define SX_F 1.0f
#define SY_F 1.0f
#define SU_F 1.0f
#define SV_F 1.0f
#define SD_F 1.0f

typedef float v2f __attribute__((ext_vector_type(2)));
typedef float v8f __attribute__((ext_vector_type(8)));

// ---------------------------------------------------------------------------
// Pass 1: z-resample as f32 WMMA GEMM.
//   volz[m, v] = sum_d vol[m, d] * Wz[v, d],   m = x*H + y  (pixel index)
// M = W*H = 36864, N = V = 48, K = D = 48 -> V_WMMA_F32_16X16X4_F32, 12 K-steps.
// One 16x16 C tile per wave32. EXEC is all-ones (no divergence, exact grid).
// ---------------------------------------------------------------------------
__device__ __forceinline__ float wz_entry(int v, int d) {
    // Reference z-interp weights (branch-free selects, keeps EXEC uniform).
    float zi  = ((float)v - (V_N - 1) * 0.5f) * (SV_F / SD_F) + (D_N - 1) * 0.5f;
    float z0f = floorf(zi);
    int   z0  = (int)z0f;
    int   z1  = z0 + 1;
    float wz  = zi - z0f;
    float m0  = (z0 >= 0 && z0 <= D_N - 1) ? 1.0f : 0.0f;
    float m1  = (z1 >= 0 && z1 <= D_N - 1) ? 1.0f : 0.0f;
    int  z0c  = min(max(z0, 0), D_N - 1);
    int  z1c  = min(max(z1, 0), D_N - 1);
    float r   = (d == z0c) ? (1.0f - wz) * m0 : 0.0f;
    r        += (d == z1c) ? wz * m1 : 0.0f;
    return r;
}

__global__ void zresample_wmma_kernel(const float* __restrict__ vol,
                                      float* __restrict__ volz) {
    const int lane = threadIdx.x & 31;
    const int wave = threadIdx.x >> 5;
    const int tile = blockIdx.x * (blockDim.x >> 5) + wave;   // 6912 tiles total
    const int ntile = tile % 3;          // N tiles: 48/16
    const int mtile = tile / 3;          // M tiles: 36864/16
    const int m_base = mtile * 16;
    const int n_base = ntile * 16;

    const int mrow = m_base + (lane & 15);       // A-matrix row (pixel)
    const int nrow = n_base + (lane & 15);       // B-matrix column (v)
    const int koff = (lane < 16) ? 0 : 2;        // ISA 16x4 A layout: K={0,1}|{2,3}

    v8f c = {};
#pragma unroll
    for (int k0 = 0; k0 < D_N; k0 += 4) {
        // A fragment: A[M=mrow, K=k0+koff .. +1]  (8B-aligned: k0+koff is even)
        const float* ap = vol + (size_t)mrow * D_N + (k0 + koff);
        v2f a;
        a.x = ap[0];
        a.y = ap[1];
        // B fragment: B[K][N] = Wz[N=v][K=d]
        v2f b;
        b.x = wz_entry(nrow, k0 + koff);
        b.y = wz_entry(nrow, k0 + koff + 1);
        c = __builtin_amdgcn_wmma_f32_16x16x4_f32(
                /*neg_a=*/false, a, /*neg_b=*/false, b,
                /*c_mod=*/(short)0, c, /*reuse_a=*/false, /*reuse_b=*/false);
    }

    // C/D layout: VGPR r -> M = r (lanes 0-15) / r+8 (lanes 16-31), N = lane%16
    const int M0 = (lane < 16) ? 0 : 8;
    const int N  = lane & 15;
#pragma unroll
    for (int r = 0; r < 8; ++r) {
        volz[(size_t)(m_base + M0 + r) * V_N + (n_base + N)] = c[r];
    }
}

// ---------------------------------------------------------------------------
// Pass 2: ray march. thread = (u, a, v-quad). v-quads fastest -> coalesced
// float4 (b128) gathers along v and coalesced float4 stores. Whole volume
// (7 MB) is L2-resident; weights shared across the 4 v's of each thread.
// ---------------------------------------------------------------------------
__global__ void joseph_fp_kernel(const float* __restrict__ volz,
                                 float* __restrict__ out,
                                 float negT, float dt) {
    const int tid = blockIdx.x * blockDim.x + threadIdx.x;
    const int vq = tid % (V_N / 4);
    const int t1 = tid / (V_N / 4);
    const int a  = t1 % A_N;
    const int u  = t1 / A_N;

    const float ang = (float)a * (float)(3.14159265358979323846 / (double)A_N);
    float sn, cs;
    sincosf(ang, &sn, &cs);

    const float uu  = ((float)u - (U_N - 1) * 0.5f) * SU_F;
    const float x0d = -uu * sn;
    const float y0d =  uu * cs;
    const float cx  = (W_N - 1) * 0.5f;
    const float cy  = (H_N - 1) * 0.5f;

    const float* __restrict__ base = volz + vq * 4;
    float4 acc = make_float4(0.f, 0.f, 0.f, 0.f);

#pragma unroll 4
    for (int s = 0; s < NSTEPS; ++s) {
        const float t  = negT + ((float)s + 0.5f) * dt;
        const float xi = fmaf(t, cs, x0d) * (1.0f / SX_F) + cx;
        const float yi = fmaf(t, sn, y0d) * (1.0f / SY_F) + cy;
        const float fx = floorf(xi);
        const float fy = floorf(yi);
        const int ix0 = (int)fx, iy0 = (int)fy;
        const float wx = xi - fx, wy = yi - fy;
        const float mx0 = (ix0     >= 0 && ix0     <= W_N - 1) ? 1.f : 0.f;
        const float mx1 = (ix0 + 1 >= 0 && ix0 + 1 <= W_N - 1) ? 1.f : 0.f;
        const float my0 = (iy0     >= 0 && iy0     <= H_N - 1) ? 1.f : 0.f;
        const float my1 = (iy0 + 1 >= 0 && iy0 + 1 <= H_N - 1) ? 1.f : 0.f;
        const int xc0 = min(max(ix0,     0), W_N - 1);
        const int xc1 = min(max(ix0 + 1, 0), W_N - 1);
        const int yc0 = min(max(iy0,     0), H_N - 1);
        const int yc1 = min(max(iy0 + 1, 0), H_N - 1);
        const float ax = (1.f - wx) * mx0;
        const float bx = wx * mx1;
        const float ay = (1.f - wy) * my0;
        const float by = wy * my1;
        const float w00 = ay * ax, w01 = ay * bx, w10 = by * ax, w11 = by * bx;

        const float4 g00 = *(const float4*)(base + (size_t)(xc0 * H_N + yc0) * V_N);
        const float4 g01 = *(const float4*)(base + (size_t)(xc1 * H_N + yc0) * V_N);
        const float4 g10 = *(const float4*)(base + (size_t)(xc0 * H_N + yc1) * V_N);
        const float4 g11 = *(const float4*)(base + (size_t)(xc1 * H_N + yc1) * V_N);

        acc.x = fmaf(g00.x, w00, acc.x); acc.y = fmaf(g00.y, w00, acc.y);
        acc.z = fmaf(g00.z, w00, acc.z); acc.w = fmaf(g00.w, w00, acc.w);
        acc.x = fmaf(g01.x, w01, acc.x); acc.y = fmaf(g01.y, w01, acc.y);
        acc.z = fmaf(g01.z, w01, acc.z); acc.w = fmaf(g01.w, w01, acc.w);
        acc.x = fmaf(g10.x, w10, acc.x); acc.y = fmaf(g10.y, w10, acc.y);
        acc.z = fmaf(g10.z, w10, acc.z); acc.w = fmaf(g10.w, w10, acc.w);
        acc.x = fmaf(g11.x, w11, acc.x); acc.y = fmaf(g11.y, w11, acc.y);
        acc.z = fmaf(g11.z, w11, acc.z); acc.w = fmaf(g11.w, w11, acc.w);
    }

    const int oidx = (u * A_N + a) * V_N + vq * 4;
    float4 r;
    r.x = acc.x * dt; r.y = acc.y * dt; r.z = acc.z * dt; r.w = acc.w * dt;
    *(float4*)(out + oidx) = r;
}

// ---------------------------------------------------------------------------
// Fallback (workspace too small): inline z-interp, scalar v per thread.
// ---------------------------------------------------------------------------
__global__ void joseph_fp_direct_kernel(const float* __restrict__ vol,
                                        float* __restrict__ out,
                                        float negT, float dt) {
    const int tid = blockIdx.x * blockDim.x + threadIdx.x;
    const int v  = tid % V_N;
    const int t1 = tid / V_N;
    const int a  = t1 % A_N;
    const int u  = t1 / A_N;

    // z weights (once per thread)
    const float zi  = ((float)v - (V_N - 1) * 0.5f) * (SV_F / SD_F) + (D_N - 1) * 0.5f;
    const float z0f = floorf(zi);
    const int   z0  = (int)z0f;
    const int   z1  = z0 + 1;
    const float wzz = zi - z0f;
    const float mz0 = (z0 >= 0 && z0 <= D_N - 1) ? 1.f : 0.f;
    const float mz1 = (z1 >= 0 && z1 <= D_N - 1) ? 1.f : 0.f;
    const int z0c = min(max(z0, 0), D_N - 1);
    const int z1c = min(max(z1, 0), D_N - 1);
    const float wlo = (1.f - wzz) * mz0;
    const float whi = wzz * mz1;

    const float ang = (float)a * (float)(3.14159265358979323846 / (double)A_N);
    float sn, cs;
    sincosf(ang, &sn, &cs);
    const float uu  = ((float)u - (U_N - 1) * 0.5f) * SU_F;
    const float x0d = -uu * sn;
    const float y0d =  uu * cs;
    const float cx  = (W_N - 1) * 0.5f;
    const float cy  = (H_N - 1) * 0.5f;

    float acc = 0.f;
#pragma unroll 2
    for (int s = 0; s < NSTEPS; ++s) {
        const float t  = negT + ((float)s + 0.5f) * dt;
        const float xi = fmaf(t, cs, x0d) * (1.0f / SX_F) + cx;
        const float yi = fmaf(t, sn, y0d) * (1.0f / SY_F) + cy;
        const float fx = floorf(xi);
        const float fy = floorf(yi);
        const int ix0 = (int)fx, iy0 = (int)fy;
        const float wx = xi - fx, wy = yi - fy;
        const float mx0 = (ix0     >= 0 && ix0     <= W_N - 1) ? 1.f : 0.f;
        const float mx1 = (ix0 + 1 >= 0 && ix0 + 1 <= W_N - 1) ? 1.f : 0.f;
        const float my0 = (iy0     >= 0 && iy0     <= H_N - 1) ? 1.f : 0.f;
        const float my1 = (iy0 + 1 >= 0 && iy0 + 1 <= H_N - 1) ? 1.f : 0.f;
        const int xc0 = min(max(ix0,     0), W_N - 1);
        const int xc1 = min(max(ix0 + 1, 0), W_N - 1);
        const int yc0 = min(max(iy0,     0), H_N - 1);
        const int yc1 = min(max(iy0 + 1, 0), H_N - 1);
        const float ax = (1.f - wx) * mx0;
        const float bx = wx * mx1;
        const float ay = (1.f - wy) * my0;
        const float by = wy * my1;

        const size_t p00 = (size_t)(xc0 * H_N + yc0) * D_N;
        const size_t p01 = (size_t)(xc1 * H_N + yc0) * D_N;
        const size_t p10 = (size_t)(xc0 * H_N + yc1) * D_N;
        const size_t p11 = (size_t)(xc1 * H_N + yc1) * D_N;
        const float g00 = vol[p00 + z0c] * wlo + vol[p00 + z1c] * whi;
        const float g01 = vol[p01 + z0c] * wlo + vol[p01 + z1c] * whi;
        const float g10 = vol[p10 + z0c] * wlo + vol[p10 + z1c] * whi;
        const float g11 = vol[p11 + z0c] * wlo + vol[p11 + z1c] * whi;

        acc = fmaf(g00, ay * ax, acc);
        acc = fmaf(g01, ay * bx, acc);
        acc = fmaf(g10, by * ax, acc);
        acc = fmaf(g11, by * bx, acc);
    }
    out[(u * A_N + a) * V_N + v] = acc * dt;
}

extern "C" void kernel_launch(void* const* d_in, const int* in_sizes, int n_in,
                              void* d_out, int out_size, void* d_ws, size_t ws_size,
                              hipStream_t stream) {
    const float* vol = (const float*)d_in[0];
    float* out = (float*)d_out;

    const double Td  = 0.5 * sqrt(((W_N - 1) * (double)SX_F) * ((W_N - 1) * (double)SX_F) +
                                  ((H_N - 1) * (double)SY_F) * ((H_N - 1) * (double)SY_F));
    const double DTd = 2.0 * Td / (double)NSTEPS;
    const float negT = (float)(-Td);
    const float dtF  = (float)DTd;

    const size_t volz_bytes = (size_t)W_N * H_N * V_N * sizeof(float);

    if (d_ws != nullptr && ws_size >= volz_bytes) {
        float* volz = (float*)d_ws;
        // Pass 1: WMMA z-resample. 36864/16 * 48/16 = 6912 tiles, 8 waves/block.
        zresample_wmma_kernel<<<6912 / 8, 256, 0, stream>>>(vol, volz);
        // Pass 2: ray march. U*A*(V/4) = 331776 threads.
        const int total = U_N * A_N * (V_N / 4);
        joseph_fp_kernel<<<total / 256, 256, 0, stream>>>(volz, out, negT, dtF);
    } else {
        const int total = U_N * A_N * V_N;  // 1327104, divisible by 256
        joseph_fp_direct_kernel<<<total / 256, 256, 0, stream>>>(vol, out, negT, dtF);
    }
}